// GCNnet_60722247631463
// MI455X (gfx1250) — compile-verified
//
#include <hip/hip_runtime.h>

typedef __attribute__((ext_vector_type(2))) float v2f;
typedef __attribute__((ext_vector_type(8))) float v8f;

// ---------------------------------------------------------------------------
// GEMM: out[M x 128] = optional_relu( A[M x K] @ W[K x 128] (+ bias) )
// fp32 WMMA (V_WMMA_F32_16X16X4_F32), wave32.
// Per ISA 7.12.2:
//   A 16x4 f32:  lanes 0-15 hold M=0..15 / K={0,1}; lanes 16-31 K={2,3}
//   B 4x16 f32:  mirrored (lanes 0-15: K rows {0,1}, lanes 16-31: {2,3})
//   C/D 16x16:   VGPR r: lanes0-15 -> M=r, N=lane; lanes16-31 -> M=r+8
// One wave computes a 16x128 tile (8 accumulators). Weights are staged into
// LDS *pre-swizzled into B-fragment layout* Wf[kstep][ntile][lane][2], so the
// inner loop is a single 8B-aligned ds_load_b64 per fragment (no repacking
// movs, conflict-free reads: lane l hits banks 2l,2l+1).
// ---------------------------------------------------------------------------
template <int K, bool RELU, bool BIAS>
__global__ __launch_bounds__(256)
void gemm_wmma_f32(const float* __restrict__ A, const float* __restrict__ W,
                   const float* __restrict__ bias, float* __restrict__ out,
                   int M)
{
    constexpr int KC = 32;                 // K-chunk per LDS stage
    constexpr int KS = KC / 4;             // 8 WMMA k-steps per chunk
    __shared__ float Wf[KS][8][32][2];     // [kstep][ntile][lane][2] = 16 KB

    const int tid   = threadIdx.x;
    const int lane  = tid & 31;
    const int wave  = tid >> 5;
    const int mbase = (blockIdx.x * 8 + wave) * 16;
    const bool active = (mbase < M);       // wave-uniform (M % 16 == 0)
    const int nl   = lane & 15;
    const int koff = (lane >> 4) << 1;     // 0 for lanes 0-15, 2 for 16-31

    v8f acc[8] = {};

    const float* Arow   = A + (size_t)(mbase + nl) * K;
    const float* WfLane = &Wf[0][0][lane][0];

    for (int k0 = 0; k0 < K; k0 += KC) {
        // cooperative stage of W[k0:k0+KC][0:128], swizzled to fragment layout
        for (int idx = tid; idx < KC * 128; idx += 256) {
            int kk = idx >> 7, n = idx & 127;          // coalesced global read
            float v = W[(size_t)(k0 + kk) * 128 + n];
            int s = kk >> 2, j = kk & 1, half = (kk >> 1) & 1;
            Wf[s][n >> 4][half * 16 + (n & 15)][j] = v;
        }
        __syncthreads();

        if (active) {
            #pragma unroll
            for (int s = 0; s < KS; ++s) {
                v2f a = *(const v2f*)(Arow + k0 + s * 4 + koff);
                #pragma unroll
                for (int t = 0; t < 8; ++t) {
                    v2f b = *(const v2f*)(WfLane + s * 512 + t * 64);
                    acc[t] = __builtin_amdgcn_wmma_f32_16x16x4_f32(
                        false, a, false, b, (short)0, acc[t], false, false);
                }
            }
        }
        __syncthreads();
    }

    if (active) {
        const int r0 = mbase + ((lane >> 4) << 3);
        #pragma unroll
        for (int t = 0; t < 8; ++t) {
            int col = t * 16 + nl;
            float bv = BIAS ? bias[col] : 0.0f;
            #pragma unroll
            for (int r = 0; r < 8; ++r) {
                float v = acc[t][r] + bv;
                if (RELU) v = fmaxf(v, 0.0f);
                out[(size_t)(r0 + r) * 128 + col] = v;
            }
        }
    }
}

// ---------------------------------------------------------------------------
// Edge / elementwise kernels
// ---------------------------------------------------------------------------
__global__ void fill_ones(float* __restrict__ p, int n) {
    int i = blockIdx.x * blockDim.x + threadIdx.x;
    if (i < n) p[i] = 1.0f;
}

__global__ void deg_count(const long long* __restrict__ dst,
                          float* __restrict__ deg, int E) {
    int i = blockIdx.x * blockDim.x + threadIdx.x;
    if (i < E) atomicAdd(&deg[dst[i]], 1.0f);
}

__global__ void to_rsqrt(float* __restrict__ p, int n) {
    int i = blockIdx.x * blockDim.x + threadIdx.x;
    if (i < n) p[i] = rsqrtf(p[i]);
}

// agg[i][c] = h[i][c] * norm[i]^2   (self-loop term), float4 per thread
__global__ void self_init(const float* __restrict__ h,
                          const float* __restrict__ norm,
                          float* __restrict__ agg, int n_nodes) {
    int i = blockIdx.x * blockDim.x + threadIdx.x;
    if (i < n_nodes * 32) {
        int node = i >> 5;
        float nv = norm[node];
        float c  = nv * nv;
        float4 v = ((const float4*)h)[i];
        float4 o;
        o.x = v.x * c; o.y = v.y * c; o.z = v.z * c; o.w = v.w * c;
        ((float4*)agg)[i] = o;
    }
}

// one wave per edge: agg[dst] += h[src] * (norm[src]*norm[dst])
__global__ void edge_scatter(const long long* __restrict__ ei,
                             const float* __restrict__ h,
                             const float* __restrict__ norm,
                             float* __restrict__ agg, int E) {
    int lane = threadIdx.x & 31;
    int wid  = (blockIdx.x * blockDim.x + threadIdx.x) >> 5;
    if (wid < E) {
        long long s = ei[wid];
        long long d = ei[(size_t)E + wid];
        float coef = norm[s] * norm[d];
        float4 v = ((const float4*)(h + (size_t)s * 128))[lane];
        float* ap = agg + (size_t)d * 128 + lane * 4;
        atomicAdd(ap + 0, v.x * coef);
        atomicAdd(ap + 1, v.y * coef);
        atomicAdd(ap + 2, v.z * coef);
        atomicAdd(ap + 3, v.w * coef);
    }
}

// x[i][c] = relu(x[i][c] + bg[c]), float4 per thread
__global__ void bias_relu(float* __restrict__ x, const float* __restrict__ bg,
                          int n_nodes) {
    int i = blockIdx.x * blockDim.x + threadIdx.x;
    if (i < n_nodes * 32) {
        int q = i & 31;
        float4 v = ((float4*)x)[i];
        float4 b = ((const float4*)bg)[q];
        v.x = fmaxf(v.x + b.x, 0.0f);
        v.y = fmaxf(v.y + b.y, 0.0f);
        v.z = fmaxf(v.z + b.z, 0.0f);
        v.w = fmaxf(v.w + b.w, 0.0f);
        ((float4*)x)[i] = v;
    }
}

// one wave per test edge: out[e] = dot(x[src], x[dst]) over 128 channels
__global__ void edge_score(const long long* __restrict__ ei,
                           const float* __restrict__ x,
                           float* __restrict__ out, int E) {
    int lane = threadIdx.x & 31;
    int wid  = (blockIdx.x * blockDim.x + threadIdx.x) >> 5;
    if (wid < E) {
        long long s = ei[wid];
        long long d = ei[(size_t)E + wid];
        float4 a = ((const float4*)(x + (size_t)s * 128))[lane];
        float4 b = ((const float4*)(x + (size_t)d * 128))[lane];
        float p = a.x * b.x + a.y * b.y + a.z * b.z + a.w * b.w;
        #pragma unroll
        for (int off = 16; off > 0; off >>= 1)
            p += __shfl_xor(p, off, 32);   // wave32 reduction
        if (lane == 0) out[wid] = p;
    }
}

// ---------------------------------------------------------------------------
// Launch
// ---------------------------------------------------------------------------
extern "C" void kernel_launch(void* const* d_in, const int* in_sizes, int n_in,
                              void* d_out, int out_size, void* d_ws, size_t ws_size,
                              hipStream_t stream) {
    const float*     xd  = (const float*)d_in[0];
    const float*     xm  = (const float*)d_in[1];
    const long long* ei  = (const long long*)d_in[2];   // (2,E) int64
    const long long* eit = (const long long*)d_in[3];   // (2,E_TEST) int64
    const float*     W1  = (const float*)d_in[4];
    const float*     b1  = (const float*)d_in[5];
    const float*     W2  = (const float*)d_in[6];
    const float*     b2  = (const float*)d_in[7];
    const float*     Wg  = (const float*)d_in[8];
    const float*     bg  = (const float*)d_in[9];
    float* out = (float*)d_out;

    const int ND = 20000, NM = 80000, N = ND + NM;
    const int E  = in_sizes[2] / 2;
    const int ET = in_sizes[3] / 2;

    char*  ws   = (char*)d_ws;
    float* X    = (float*)ws;                               // N*128 f32
    float* H    = (float*)(ws + (size_t)N * 128 * 4);       // N*128 f32
    float* norm = (float*)(ws + (size_t)N * 128 * 4 * 2);   // N f32

    // MLPs (relu + bias), writing into concat buffer X
    gemm_wmma_f32<256, true, true><<<(ND + 127) / 128, 256, 0, stream>>>(
        xd, W1, b1, X, ND);
    gemm_wmma_f32<256, true, true><<<(NM + 127) / 128, 256, 0, stream>>>(
        xm, W2, b2, X + (size_t)ND * 128, NM);

    // degree -> rsqrt norm (deg starts at 1 for self loop)
    fill_ones<<<(N + 255) / 256, 256, 0, stream>>>(norm, N);
    deg_count<<<(E + 255) / 256, 256, 0, stream>>>(ei + E, norm, E);
    to_rsqrt<<<(N + 255) / 256, 256, 0, stream>>>(norm, N);

    // h = X @ Wg (no bias, no relu)
    gemm_wmma_f32<128, false, false><<<(N + 127) / 128, 256, 0, stream>>>(
        X, Wg, nullptr, H, N);

    // X reused as agg: self-loop term, then scatter-add over edges
    self_init<<<(N * 32 + 255) / 256, 256, 0, stream>>>(H, norm, X, N);
    edge_scatter<<<(E * 32 + 255) / 256, 256, 0, stream>>>(ei, H, norm, X, E);

    // relu(agg + bg), then dot-product scoring of test edges
    bias_relu<<<(N * 32 + 255) / 256, 256, 0, stream>>>(X, bg, N);
    edge_score<<<(ET * 32 + 255) / 256, 256, 0, stream>>>(eit, X, out, ET);
}